// NonRigidAlignmentRefiner_27504970564304
// MI455X (gfx1250) — compile-verified
//
#include <hip/hip_runtime.h>
#include <math.h>

// ---------------------------------------------------------------------------
// TPS non-rigid alignment for MI455X (gfx1250, wave32).
//   k1: assemble 259x261 augmented system (L | rhs) from dst = src + disp
//   k2: single-workgroup Gauss-Jordan with partial pivoting (diag of K is 0!)
//   k3: per-pixel TPS warp via V_WMMA_F32_16X16X4_F32 accumulation + bilinear
//       zero-padded sampling. Image (4MB) is L2-resident; phase is VALU/trans
//       bound (~256M v_log_f32), WMMA offloads the K@Wk reduction FMAs.
//       A-operand (transposed kernel weights) is pre-laid-out in LDS per
//       (center-block, lane) so the hot loop has zero divergence/selects.
// ---------------------------------------------------------------------------

#define HH 1024
#define WW 1024
#define NPTS 256
#define NA 259          // N + 3
#define NC 261          // N + 3 + 2 rhs columns
#define LDA 264         // padded row stride (floats)
#define TPS_EPS 1e-8f
#define C_HALF_LN2 0.34657359027997264f   // 0.5 * ln(2): 0.5*d2*ln(x) = d2*C*log2(x)

typedef float v2f __attribute__((ext_vector_type(2)));
typedef float v8f __attribute__((ext_vector_type(8)));

__device__ __forceinline__ float tps_uval(float d2) {
    return (d2 * C_HALF_LN2) * __log2f(d2 + TPS_EPS);
}

// ---------------- kernel 1: build augmented system --------------------------
__global__ void __launch_bounds__(256)
tps_build(const float* __restrict__ srcp, const float* __restrict__ disp,
          float* __restrict__ A) {
    int idx = blockIdx.x * blockDim.x + threadIdx.x;
    if (idx >= NA * LDA) return;
    int r = idx / LDA;
    int j = idx - r * LDA;
    float v = 0.0f;
    if (j < NPTS) {
        float djx = srcp[2 * j]     + disp[2 * j];
        float djy = srcp[2 * j + 1] + disp[2 * j + 1];
        if (r < NPTS) {
            float drx = srcp[2 * r]     + disp[2 * r];
            float dry = srcp[2 * r + 1] + disp[2 * r + 1];
            float dx = drx - djx, dy = dry - djy;
            float d2 = dx * dx + dy * dy;
            v = tps_uval(d2);                 // 0 on diagonal
        } else if (r == NPTS)     v = 1.0f;   // P^T row of ones
        else if (r == NPTS + 1)   v = djx;    // P^T row of dst x
        else if (r == NPTS + 2)   v = djy;    // P^T row of dst y
    } else if (r < NPTS) {
        if      (j == NPTS)     v = 1.0f;                       // P col 0
        else if (j == NPTS + 1) v = srcp[2 * r]     + disp[2 * r];      // dst x
        else if (j == NPTS + 2) v = srcp[2 * r + 1] + disp[2 * r + 1];  // dst y
        else if (j == NPTS + 3) v = srcp[2 * r];                // rhs x = src x
        else if (j == NPTS + 4) v = srcp[2 * r + 1];            // rhs y = src y
        // j >= NC : padding stays 0
    }
    A[idx] = v;
}

// ---------------- kernel 2: Gauss-Jordan with partial pivoting --------------
__global__ void __launch_bounds__(1024)
tps_solve(float* __restrict__ A, float* __restrict__ sol) {
    __shared__ float piv[NC];
    __shared__ float fac[NA];
    __shared__ float redv[32];
    __shared__ int   redr[32];
    __shared__ int   spiv;
    const int t = threadIdx.x;
    const int lane = t & 31;
    const int wv = t >> 5;

    for (int k = 0; k < NA; ++k) {
        // argmax |A[r][k]| over r in [k, NA)
        float bv = -1.0f;
        int br = k;
        for (int r = k + t; r < NA; r += 1024) {
            float a = fabsf(A[r * LDA + k]);
            if (a > bv) { bv = a; br = r; }
        }
        for (int off = 16; off > 0; off >>= 1) {
            float ov = __shfl_down(bv, off);
            int   orr = __shfl_down(br, off);
            if (ov > bv) { bv = ov; br = orr; }
        }
        if (lane == 0) { redv[wv] = bv; redr[wv] = br; }
        __syncthreads();
        if (wv == 0) {
            bv = redv[lane];
            br = redr[lane];
            for (int off = 16; off > 0; off >>= 1) {
                float ov = __shfl_down(bv, off);
                int   orr = __shfl_down(br, off);
                if (ov > bv) { bv = ov; br = orr; }
            }
            if (lane == 0) spiv = br;
        }
        __syncthreads();

        const int pr = spiv;
        if (pr != k) {  // uniform branch
            for (int j = t; j < NC; j += 1024) {
                float a = A[k * LDA + j];
                float b = A[pr * LDA + j];
                A[k * LDA + j] = b;
                A[pr * LDA + j] = a;
            }
        }
        __syncthreads();

        const float pvv = A[k * LDA + k];   // all threads read pivot first
        __syncthreads();                    // ...before anyone normalizes it
        const float inv = 1.0f / pvv;
        for (int j = t; j < NC; j += 1024) {
            float v = A[k * LDA + j] * inv;
            piv[j] = v;
            A[k * LDA + j] = v;
        }
        for (int r = t; r < NA; r += 1024) fac[r] = A[r * LDA + k]; // fac[k] unused
        __syncthreads();

        // eliminate all rows r != k; cols < k are already unit columns
        for (int r = wv; r < NA; r += 32) {
            if (r == k) continue;
            const float f = fac[r];
            for (int j = k + lane; j < NC; j += 32)
                A[r * LDA + j] -= f * piv[j];
        }
        __syncthreads();
    }

    // solution sits in the two augmented columns
    for (int i = t; i < NA; i += 1024) {
        sol[2 * i]     = A[i * LDA + NA];
        sol[2 * i + 1] = A[i * LDA + NA + 1];
    }
}

// ---------------- kernel 3: warp grid via WMMA + bilinear sample ------------
// Per wave: two 16-pixel tiles. For each 4-center block cb:
//   A (16x4)  = Wk^T slice: row m (=lane&15) is output comp (0:x, 1:y, rest 0),
//               K pair {2h, 2h+1}. Values depend only on (cb, lane) -> fully
//               precomputed into sA[cb][lane] (16KB LDS), one ds_load_b64/iter.
//   B (4x16)  = U: lane computes u(pixel=lane&15, centers 4cb+2h, 4cb+2h+1).
//   D (16x16) : row0 = x-warp, row1 = y-warp for the 16 pixel columns.
__global__ void __launch_bounds__(256)
tps_warp_sample(const float* __restrict__ img, const float* __restrict__ srcp,
                const float* __restrict__ sol, float* __restrict__ out) {
    __shared__ __align__(16) float scw[NPTS * 2];       // centers (cx,cy)
    __shared__ __align__(16) float sA[64 * 32 * 2];     // A pairs per (cb, lane)
    __shared__ float saff[6];                           // affine rows 256..258
    const int t = threadIdx.x;
    for (int i = t; i < NPTS * 2; i += 256) scw[i] = srcp[i];
    // Pre-layout the WMMA A operand: sA[cb*32+l] = (w[4cb+2h][j], w[4cb+2h+1][j])
    for (int idx = t; idx < 64 * 32; idx += 256) {
        const int cb = idx >> 5;
        const int l  = idx & 31;
        const int j  = l & 15;
        const int c0 = 4 * cb + 2 * (l >> 4);
        float a0 = 0.0f, a1 = 0.0f;
        if (j < 2) { a0 = sol[2 * c0 + j]; a1 = sol[2 * c0 + 2 + j]; }
        sA[2 * idx]     = a0;
        sA[2 * idx + 1] = a1;
    }
    if (t < 6) saff[t] = sol[NPTS * 2 + t];
    __syncthreads();

    const int lid = t & 31;
    const int col = lid & 15;
    const int h = lid >> 4;
    const int base = blockIdx.x * 256 + (t >> 5) * 32;  // wave's first pixel
    const int xbase = base & (WW - 1);
    const int yrow = base >> 10;
    const float s = 2.0f / (float)(WW - 1);
    const float dx16 = -16.0f * s;                 // dxB = dxA + gx step
    const float gy = yrow * s - 1.0f;
    const float gxa = (xbase + col) * s - 1.0f;    // tile A pixel column

    v8f accA = {};
    v8f accB = {};
    const float* ccp = &scw[(2 * h) * 2];          // + 8 floats per iteration
    const float* ap  = &sA[lid * 2];               // +64 floats per iteration
    #pragma unroll 8
    for (int cb = 0; cb < 64; ++cb, ccp += 8, ap += 64) {
        const float4 cc = *reinterpret_cast<const float4*>(ccp); // 16B aligned
        const v2f a     = *reinterpret_cast<const v2f*>(ap);     // 8B aligned
        // shared dy^2 per center; tile-B dx is tile-A dx plus a constant
        const float dy0 = gy - cc.y;  const float dy20 = dy0 * dy0;
        const float dy1 = gy - cc.w;  const float dy21 = dy1 * dy1;
        const float dxa0 = gxa - cc.x, dxb0 = dxa0 - dx16;
        const float dxa1 = gxa - cc.z, dxb1 = dxa1 - dx16;
        v2f bA, bB;
        bA.x = tps_uval(fmaf(dxa0, dxa0, dy20));
        bA.y = tps_uval(fmaf(dxa1, dxa1, dy21));
        bB.x = tps_uval(fmaf(dxb0, dxb0, dy20));
        bB.y = tps_uval(fmaf(dxb1, dxb1, dy21));
        accA = __builtin_amdgcn_wmma_f32_16x16x4_f32(false, a, false, bA,
                                                     (short)0, accA, false, false);
        accB = __builtin_amdgcn_wmma_f32_16x16x4_f32(false, a, false, bB,
                                                     (short)0, accB, false, false);
    }

    // D rows 0/1 live in acc[0]/acc[1] on lanes 0-15; pull tile B to hi half
    float bx = __shfl(accB[0], col, 32);
    float by = __shfl(accB[1], col, 32);
    float kx = (lid < 16) ? accA[0] : bx;
    float ky = (lid < 16) ? accA[1] : by;

    const float gxo = (xbase + lid) * s - 1.0f;
    const float wxp = kx + saff[0] + gxo * saff[2] + gy * saff[4];
    const float wyp = ky + saff[1] + gxo * saff[3] + gy * saff[5];

    // bilinear, zeros padding: x = ((wx+1)*W - 1) * 0.5
    const float fx = (wxp + 1.0f) * (0.5f * (float)WW) - 0.5f;
    const float fy = (wyp + 1.0f) * (0.5f * (float)HH) - 0.5f;
    const float x0f = floorf(fx), y0f = floorf(fy);
    const float tx = fx - x0f, ty = fy - y0f;
    const int x0 = (int)x0f, y0 = (int)y0f;
    const int x1 = x0 + 1, y1 = y0 + 1;

    auto samp = [&](int xi, int yi) -> float {
        bool in = (xi >= 0) & (xi < WW) & (yi >= 0) & (yi < HH);
        int xc = min(max(xi, 0), WW - 1);
        int yc = min(max(yi, 0), HH - 1);
        float v = img[yc * WW + xc];
        return in ? v : 0.0f;
    };
    const float v00 = samp(x0, y0);
    const float v10 = samp(x1, y0);
    const float v01 = samp(x0, y1);
    const float v11 = samp(x1, y1);
    const float r0 = v00 * (1.0f - tx) + v10 * tx;
    const float r1 = v01 * (1.0f - tx) + v11 * tx;
    out[base + lid] = r0 * (1.0f - ty) + r1 * ty;
}

// ---------------------------------------------------------------------------
extern "C" void kernel_launch(void* const* d_in, const int* in_sizes, int n_in,
                              void* d_out, int out_size, void* d_ws, size_t ws_size,
                              hipStream_t stream) {
    const float* img  = (const float*)d_in[0];   // (1,1,1024,1024)
    const float* srcp = (const float*)d_in[1];   // (256,2)
    const float* disp = (const float*)d_in[2];   // (256,2)
    float* A   = (float*)d_ws;                   // 259 x 264 augmented matrix
    float* sol = A + NA * LDA;                   // 259 x 2 solution
    float* out = (float*)d_out;                  // (1024,1024)

    tps_build<<<(NA * LDA + 255) / 256, 256, 0, stream>>>(srcp, disp, A);
    tps_solve<<<1, 1024, 0, stream>>>(A, sol);
    tps_warp_sample<<<(HH * WW) / 256, 256, 0, stream>>>(img, srcp, sol, out);
}